// TransformerBlock_1262720385537
// MI455X (gfx1250) — compile-verified
//
#include <hip/hip_runtime.h>
#include <math.h>

// ---------------------------------------------------------------------------
// MI455X (gfx1250) transformer decode block, fp32.
// Roofline: dominated by the 4.3 GiB cache passthrough (~185us @ 23.3 TB/s);
// GEMMs are weight-bandwidth-bound (200 MB), so weights stay fp32 and use
// V_WMMA_F32_16X16X4_F32. This revision batches A/B fragment loads into
// register arrays so the scheduler emits deep load clauses with countdown
// waits (instead of load -> wait(0) -> wmma serialization), while keeping
// 4 independent accumulators for matrix-pipe ILP.
// ---------------------------------------------------------------------------

typedef float v2f __attribute__((ext_vector_type(2)));
typedef float v4f __attribute__((ext_vector_type(4)));
typedef float v8f __attribute__((ext_vector_type(8)));

#define BATCH   128
#define N_HEADS 16
#define SEQ_L   1024
#define D_HEAD  128
#define D_MODEL 2048
#define FFN     8192

// ---------------------------------------------------------------------------
// LayerNorm: one 256-thread block per row (128 rows, D=2048 -> 8 elems/thread)
// ---------------------------------------------------------------------------
__global__ __launch_bounds__(256) void ln_kernel(const float* __restrict__ x,
                                                 const float* __restrict__ scale,
                                                 const float* __restrict__ bias,
                                                 float* __restrict__ out)
{
    __shared__ float red[256];
    const int row = blockIdx.x;
    const int tid = threadIdx.x;
    const float* xr = x + (size_t)row * D_MODEL;

    float loc[8];
    float s = 0.f;
#pragma unroll
    for (int i = 0; i < 8; ++i) { loc[i] = xr[tid + i * 256]; s += loc[i]; }
    red[tid] = s; __syncthreads();
    for (int o = 128; o >= 1; o >>= 1) {
        if (tid < o) red[tid] += red[tid + o];
        __syncthreads();
    }
    const float mu = red[0] * (1.f / D_MODEL);
    __syncthreads();

    float v = 0.f;
#pragma unroll
    for (int i = 0; i < 8; ++i) { float d = loc[i] - mu; v += d * d; }
    red[tid] = v; __syncthreads();
    for (int o = 128; o >= 1; o >>= 1) {
        if (tid < o) red[tid] += red[tid + o];
        __syncthreads();
    }
    const float inv = rsqrtf(red[0] * (1.f / D_MODEL) + 1e-6f);

    float* orow = out + (size_t)row * D_MODEL;
#pragma unroll
    for (int i = 0; i < 8; ++i) {
        const int c = tid + i * 256;
        orow[c] = (loc[i] - mu) * inv * scale[c] + bias[c];
    }
}

// ---------------------------------------------------------------------------
// GEMM: out(128,N) = A(128,K) @ B(K,N)  [+bias] [+resid | gelu]
// 8 waves/block; wave w computes the 16x16 tile at rows [16w,16w+16).
// Per K-chunk of 64:
//   stage B 64x16 in LDS (4 NT global loads -> 4 DS stores, split loops),
//   preload 16 A frags (global, claused b64) + 16 B frags (DS, claused)
//   into register arrays, then run 16 WMMAs rotating over 4 accumulators.
// WMMA f32 16x16x4 ISA layout:
//   A frag (v2f): lane<16 -> A[m0+l][k..k+1]; lane>=16 -> k+2..k+3
//   B frag (v2f): lane<16 -> B[k..k+1][n0+l]; lane>=16 -> k+2..k+3
// op: 0 = plain, 1 = +bias +resid, 2 = +bias then tanh-GELU
// ---------------------------------------------------------------------------
__global__ __launch_bounds__(256) void gemm128_wmma_f32(
    const float* __restrict__ A, const float* __restrict__ B,
    const float* __restrict__ bias, const float* __restrict__ resid,
    float* __restrict__ out, int K, int N, int op)
{
    __shared__ float bs[64][16];

    const int n0   = blockIdx.x * 16;
    const int tid  = threadIdx.x;
    const int lane = tid & 31;
    const int wave = tid >> 5;
    const int m0   = wave * 16;
    const int half = lane >> 4;
    const int l16  = lane & 15;

    v8f acc[4];
#pragma unroll
    for (int i = 0; i < 4; ++i) acc[i] = (v8f){0.f,0.f,0.f,0.f,0.f,0.f,0.f,0.f};

    const float* arow = A + (size_t)(m0 + l16) * K + half * 2;
    // B staging: thread t covers (kk = t>>4 + 16i, nn = t&15) of each chunk
    const int skk = tid >> 4;
    const int snn = tid & 15;
    const float* bptr = B + (size_t)skk * N + n0 + snn;

    for (int k0 = 0; k0 < K; k0 += 64) {
        float tmp[4];
#pragma unroll
        for (int i = 0; i < 4; ++i)
            tmp[i] = __builtin_nontemporal_load(bptr + (size_t)(16 * i) * N);
        bptr += (size_t)64 * N;

        __syncthreads();            // previous chunk's DS reads are done
#pragma unroll
        for (int i = 0; i < 4; ++i)
            bs[skk + 16 * i][snn] = tmp[i];
        __syncthreads();            // chunk visible to all waves

        v2f af[16], bf[16];
#pragma unroll
        for (int i = 0; i < 16; ++i)          // 16-deep global b64 clause
            af[i] = *(const v2f*)(arow + i * 4);
#pragma unroll
        for (int i = 0; i < 16; ++i) {        // 16-deep DS clause
            bf[i].x = bs[i * 4 + half * 2 + 0][l16];
            bf[i].y = bs[i * 4 + half * 2 + 1][l16];
        }
        arow += 64;

#pragma unroll
        for (int i = 0; i < 16; ++i)          // countdown waits, 4-way ILP
            acc[i & 3] = __builtin_amdgcn_wmma_f32_16x16x4_f32(
                false, af[i], false, bf[i], (short)0, acc[i & 3], false, false);
    }

    v8f accs = (acc[0] + acc[1]) + (acc[2] + acc[3]);

    // C/D layout: VGPR r -> (M = m0 + half*8 + r, N = n0 + l16)
    const int col   = n0 + l16;
    const int rbase = m0 + half * 8;
    const float bv  = (op != 0) ? bias[col] : 0.f;
#pragma unroll
    for (int r = 0; r < 8; ++r) {
        float v = accs[r] + bv;
        const size_t off = (size_t)(rbase + r) * N + col;
        if (op == 2) {
            const float x3 = v * v * v;
            v = 0.5f * v * (1.f + tanhf(0.7978845608028654f * (v + 0.044715f * x3)));
        } else if (op == 1) {
            v += resid[off];
        }
        out[off] = v;
    }
}

// ---------------------------------------------------------------------------
// Scatter new k/v row into the cache copies in d_out at position idx.
// ---------------------------------------------------------------------------
__global__ __launch_bounds__(128) void update_cache_kernel(
    const float* __restrict__ qkv, float* __restrict__ out_k,
    float* __restrict__ out_v, const int* __restrict__ idx_p)
{
    const int idx = *idx_p;
    const int bh  = blockIdx.x;              // b*16 + h
    const int b   = bh >> 4;
    const int h   = bh & 15;
    const int d   = threadIdx.x;
    const size_t co = ((size_t)bh * SEQ_L + idx) * D_HEAD + d;
    const size_t qo = (size_t)b * (3 * D_MODEL) + h * D_HEAD + d;
    out_k[co] = qkv[qo + D_MODEL];
    out_v[co] = qkv[qo + 2 * D_MODEL];
}

// ---------------------------------------------------------------------------
// Attention: one block per (b,h). Scores via wave-per-position dot products
// (16B coalesced non-temporal K reads, shuffle reduce), softmax in LDS, then
// o[d] via coalesced V column accumulation. Position idx uses the freshly
// computed k/v from the QKV buffer (input caches are never modified).
// ---------------------------------------------------------------------------
__global__ __launch_bounds__(256) void attn_kernel(
    const float* __restrict__ qkv,
    const float* __restrict__ cache_k,
    const float* __restrict__ cache_v,
    float* __restrict__ o,
    const int* __restrict__ idx_p)
{
    __shared__ float sc[SEQ_L];
    __shared__ float red[256];

    const int idx  = *idx_p;
    const int Lkv  = idx + 1;
    const int b    = blockIdx.x >> 4;
    const int h    = blockIdx.x & (N_HEADS - 1);
    const int lane = threadIdx.x & 31;
    const int wave = threadIdx.x >> 5;
    const int tid  = threadIdx.x;

    const float* qrow = qkv + (size_t)b * (3 * D_MODEL) + h * D_HEAD;
    const float* knew = qrow + D_MODEL;
    const float* vnew = qrow + 2 * D_MODEL;
    const v4f q4      = *(const v4f*)(qrow + lane * 4);
    const float scale = 0.08838834764831845f;   // 1/sqrt(128)

    const float* Kb = cache_k + (size_t)blockIdx.x * SEQ_L * D_HEAD;
    for (int p = wave; p < Lkv; p += 8) {
        const float* krow = (p == idx) ? knew : (Kb + (size_t)p * D_HEAD);
        const v4f k4 = __builtin_nontemporal_load((const v4f*)krow + lane);
        float part = q4.x * k4.x + q4.y * k4.y + q4.z * k4.z + q4.w * k4.w;
#pragma unroll
        for (int off = 16; off >= 1; off >>= 1)
            part += __shfl_down(part, off, 32);
        if (lane == 0) sc[p] = part * scale;
    }
    __syncthreads();

    // softmax over sc[0..Lkv)
    float m = -3.402823466e38f;
    for (int p = tid; p < Lkv; p += 256) m = fmaxf(m, sc[p]);
    red[tid] = m; __syncthreads();
    for (int off = 128; off >= 1; off >>= 1) {
        if (tid < off) red[tid] = fmaxf(red[tid], red[tid + off]);
        __syncthreads();
    }
    m = red[0];
    __syncthreads();

    float ssum = 0.f;
    for (int p = tid; p < Lkv; p += 256) {
        const float e = __expf(sc[p] - m);
        sc[p] = e;
        ssum += e;
    }
    red[tid] = ssum; __syncthreads();
    for (int off = 128; off >= 1; off >>= 1) {
        if (tid < off) red[tid] += red[tid + off];
        __syncthreads();
    }
    const float inv = 1.f / red[0];
    __syncthreads();

    // o[d] = sum_p attn[p] * V[p][d]; two thread-groups split p, combine in LDS
    const int d = tid & (D_HEAD - 1);
    const int g = tid >> 7;
    const float* Vb = cache_v + (size_t)blockIdx.x * SEQ_L * D_HEAD;
    float accv = 0.f;
    for (int p = g; p < Lkv; p += 2) {
        const float* vrow = (p == idx) ? vnew : (Vb + (size_t)p * D_HEAD);
        accv += sc[p] * __builtin_nontemporal_load(vrow + d);
    }
    red[tid] = accv; __syncthreads();
    if (g == 0)
        o[(size_t)b * D_MODEL + h * D_HEAD + d] = (red[tid] + red[tid + 128]) * inv;
}

// ---------------------------------------------------------------------------
// Launch
// ---------------------------------------------------------------------------
extern "C" void kernel_launch(void* const* d_in, const int* in_sizes, int n_in,
                              void* d_out, int out_size, void* d_ws, size_t ws_size,
                              hipStream_t stream)
{
    const float* x       = (const float*)d_in[0];
    const float* cache_k = (const float*)d_in[1];
    const float* cache_v = (const float*)d_in[2];
    const float* ln1_s   = (const float*)d_in[3];
    const float* ln1_b   = (const float*)d_in[4];
    const float* w_qkv   = (const float*)d_in[5];
    const float* w_proj  = (const float*)d_in[6];
    const float* b_proj  = (const float*)d_in[7];
    const float* ln2_s   = (const float*)d_in[8];
    const float* ln2_b   = (const float*)d_in[9];
    const float* w_ff1   = (const float*)d_in[10];
    const float* b_ff1   = (const float*)d_in[11];
    const float* w_ff2   = (const float*)d_in[12];
    const float* b_ff2   = (const float*)d_in[13];
    const int*   idx_p   = (const int*)d_in[14];

    float* y = (float*)d_out;
    const size_t CACHE_ELEMS = (size_t)BATCH * N_HEADS * SEQ_L * D_HEAD;
    float* out_k = y + (size_t)BATCH * D_MODEL;
    float* out_v = out_k + CACHE_ELEMS;

    char* ws = (char*)d_ws;
    float* h   = (float*)(ws);                        // 1 MB (128x2048)
    float* qkv = (float*)(ws + ((size_t)1 << 20));    // 3 MB (128x6144)
    float* o   = (float*)(ws + ((size_t)4 << 20));    // 1 MB
    float* x2  = (float*)(ws + ((size_t)5 << 20));    // 1 MB
    float* h2  = (float*)(ws + ((size_t)6 << 20));    // 1 MB
    float* ffa = (float*)(ws + ((size_t)7 << 20));    // 4 MB (128x8192)

    // Cache passthrough: bulk D2D copy, idx row patched after QKV GEMM.
    hipMemcpyAsync(out_k, cache_k, CACHE_ELEMS * sizeof(float),
                   hipMemcpyDeviceToDevice, stream);
    hipMemcpyAsync(out_v, cache_v, CACHE_ELEMS * sizeof(float),
                   hipMemcpyDeviceToDevice, stream);

    ln_kernel<<<BATCH, 256, 0, stream>>>(x, ln1_s, ln1_b, h);

    gemm128_wmma_f32<<<(3 * D_MODEL) / 16, 256, 0, stream>>>(
        h, w_qkv, nullptr, nullptr, qkv, D_MODEL, 3 * D_MODEL, 0);

    update_cache_kernel<<<BATCH * N_HEADS, D_HEAD, 0, stream>>>(
        qkv, out_k, out_v, idx_p);

    attn_kernel<<<BATCH * N_HEADS, 256, 0, stream>>>(
        qkv, cache_k, cache_v, o, idx_p);

    gemm128_wmma_f32<<<D_MODEL / 16, 256, 0, stream>>>(
        o, w_proj, b_proj, x, x2, D_MODEL, D_MODEL, 1);

    ln_kernel<<<BATCH, 256, 0, stream>>>(x2, ln2_s, ln2_b, h2);

    gemm128_wmma_f32<<<FFN / 16, 256, 0, stream>>>(
        h2, w_ff1, b_ff1, nullptr, ffa, D_MODEL, FFN, 2);

    gemm128_wmma_f32<<<D_MODEL / 16, 256, 0, stream>>>(
        ffa, w_ff2, b_ff2, x2, y, FFN, D_MODEL, 1);
}